// LigerCosineLoss_51376398794763
// MI455X (gfx1250) — compile-verified
//
#include <hip/hip_runtime.h>
#include <math.h>

// Problem constants (match reference)
#define BT     4096
#define HS     2048          // student hidden
#define HT     4096          // teacher hidden
#define VOCAB  32000
#define IGNORE_INDEX (-100)
#define EPSN   1e-12f

// Tiling
#define BM  64               // rows per workgroup
#define BN  128              // vocab cols per workgroup
#define KC  64               // K chunk staged in LDS
#define LDA 68               // padded LDS row stride (floats) -> no bank conflicts
#define NBLK (VOCAB / BN)    // 250 column blocks

#define ABUF (BM * LDA)      // 4352 floats per A stage
#define BBUF (BN * LDA)      // 8704 floats per B stage
#define BUFSZ (ABUF + BBUF)  // 13056 floats per stage buffer
#define SMEM_FLOATS (2 * BUFSZ)          // 26112 floats (double buffered) >= 16384 stats
#define SMEM_BYTES  (SMEM_FLOATS * 4)    // 104448 B dynamic LDS (<320KB/WGP)

typedef __attribute__((ext_vector_type(2))) float v2f;
typedef __attribute__((ext_vector_type(8))) float v8f;

#if defined(__AMDGCN__) && \
    __has_builtin(__builtin_amdgcn_global_load_async_to_lds_b128) && \
    __has_builtin(__builtin_amdgcn_s_wait_asynccnt)
#define USE_ASYNC 1
// Builtin signature (decoded from hipcc diagnostics):
//   void __builtin_amdgcn_global_load_async_to_lds_b128(
//        AS1 v4i* gsrc, AS3 v4i* ldst, imm int offset, imm int cpol)
typedef int v4i __attribute__((vector_size(4 * sizeof(int))));
typedef __attribute__((address_space(1))) v4i g_v4i;   // global
typedef __attribute__((address_space(3))) v4i l_v4i;   // LDS
#else
#define USE_ASYNC 0
#endif

// ---------------------------------------------------------------------------
// Stage one 64xKC A tile + 128xKC B tile into an LDS buffer.
// Async path: GLOBAL_LOAD_ASYNC_TO_LDS_B128 (12 issues per wave, ASYNCcnt).
// ---------------------------------------------------------------------------
__device__ __forceinline__ void stage_tiles(
    const float* __restrict__ X, const float* __restrict__ W, int K,
    int rowBase, int colBase, int kc0,
    float* __restrict__ As, float* __restrict__ Bs, int tid)
{
#if USE_ASYNC
    #pragma unroll
    for (int i = 0; i < 4; ++i) {
        int idx = tid + i * 256;
        int r = idx >> 4, kq = idx & 15;
        __builtin_amdgcn_global_load_async_to_lds_b128(
            (g_v4i*)(X + (size_t)(rowBase + r) * K + kc0 + kq * 4),
            (l_v4i*)(As + r * LDA + kq * 4), 0, 0);
    }
    #pragma unroll
    for (int i = 0; i < 8; ++i) {
        int idx = tid + i * 256;
        int r = idx >> 4, kq = idx & 15;
        __builtin_amdgcn_global_load_async_to_lds_b128(
            (g_v4i*)(W + (size_t)(colBase + r) * K + kc0 + kq * 4),
            (l_v4i*)(Bs + r * LDA + kq * 4), 0, 0);
    }
#else
    #pragma unroll
    for (int i = 0; i < 4; ++i) {
        int idx = tid + i * 256;
        int r = idx >> 4, kq = idx & 15;
        float4 v = *(const float4*)&X[(size_t)(rowBase + r) * K + kc0 + kq * 4];
        *(float4*)&As[r * LDA + kq * 4] = v;
    }
    #pragma unroll
    for (int i = 0; i < 8; ++i) {
        int idx = tid + i * 256;
        int r = idx >> 4, kq = idx & 15;
        float4 v = *(const float4*)&W[(size_t)(colBase + r) * K + kc0 + kq * 4];
        *(float4*)&Bs[r * LDA + kq * 4] = v;
    }
#endif
}

// ---------------------------------------------------------------------------
// Double-buffered fused GEMM accumulate:
//   acc += X[rowBase:+64, :K] * W[colBase:+128, :K]^T
// V_WMMA_F32_16X16X4_F32, wave32; async staging overlapped with compute.
// ---------------------------------------------------------------------------
__device__ __forceinline__ void gemm_accum(
    const float* __restrict__ X, const float* __restrict__ W, int K,
    int rowBase, int colBase, float* __restrict__ smem, v8f acc[2][2])
{
    const int tid  = threadIdx.x;
    const int lane = tid & 31;
    const int wid  = tid >> 5;
    const int wm   = wid >> 2;        // wave row 0..1  (32-row strip)
    const int wn   = wid & 3;         // wave col 0..3  (32-col strip)
    const int half = lane >> 4;       // K-half select per ISA A/B layout
    const int mr   = lane & 15;       // row (A) / col (B) within 16

    float* const buf0 = smem;
    float* const buf1 = smem + BUFSZ;
    const int nch = K / KC;

    // Prologue: stage chunk 0 into buf0.
    stage_tiles(X, W, K, rowBase, colBase, 0, buf0, buf0 + ABUF, tid);

    for (int c = 0; c < nch; ++c) {
        float* As = (c & 1) ? buf1 : buf0;
        float* Bs = As + ABUF;

        // Issue next chunk into the alternate buffer (its last readers
        // passed the barrier at the end of iteration c-1), then wait only
        // for the *oldest* 12 async ops (this chunk) before computing.
        if (c + 1 < nch) {
            float* An = (c & 1) ? buf0 : buf1;
            stage_tiles(X, W, K, rowBase, colBase, (c + 1) * KC, An, An + ABUF, tid);
#if USE_ASYNC
            __builtin_amdgcn_s_wait_asynccnt(12);
#endif
        } else {
#if USE_ASYNC
            __builtin_amdgcn_s_wait_asynccnt(0);
#endif
        }
        // Far-ahead weight prefetch (global_prefetch_b8).
        if (c + 2 < nch) {
            const float* p = &W[(size_t)(colBase + (tid >> 1)) * K
                                + (c + 2) * KC + (tid & 1) * 32];
            __builtin_prefetch(p, 0, 1);
        }
        __syncthreads();

        const float* Ar0 = &As[(wm * 32 + mr) * LDA];
        const float* Ar1 = Ar0 + 16 * LDA;
        const float* Br0 = &Bs[(wn * 32 + mr) * LDA];
        const float* Br1 = Br0 + 16 * LDA;

        #pragma unroll 4
        for (int kk = 0; kk < KC; kk += 4) {
            const int ko = kk + 2 * half; // lanes 0-15: K{kk,kk+1}; 16-31: K{kk+2,kk+3}
            v2f a0 = { Ar0[ko], Ar0[ko + 1] };
            v2f a1 = { Ar1[ko], Ar1[ko + 1] };
            v2f b0 = { Br0[ko], Br0[ko + 1] };
            v2f b1 = { Br1[ko], Br1[ko + 1] };
            acc[0][0] = __builtin_amdgcn_wmma_f32_16x16x4_f32(false, a0, false, b0,
                         (short)0, acc[0][0], false, false);
            acc[0][1] = __builtin_amdgcn_wmma_f32_16x16x4_f32(false, a0, false, b1,
                         (short)0, acc[0][1], false, false);
            acc[1][0] = __builtin_amdgcn_wmma_f32_16x16x4_f32(false, a1, false, b0,
                         (short)0, acc[1][0], false, false);
            acc[1][1] = __builtin_amdgcn_wmma_f32_16x16x4_f32(false, a1, false, b1,
                         (short)0, acc[1][1], false, false);
        }
        __syncthreads();
    }
}

// ---------------------------------------------------------------------------
// Kernel 1: fused student+teacher GEMM tile + per-(row, col-block) stats.
// partial[(row*NBLK + nb)*6] = { max_s, sum_exp(s-max), sum s^2, sum t^2,
//                                sum s*t, target_logit_if_in_block }
// ---------------------------------------------------------------------------
__global__ __launch_bounds__(256) void k_fused_gemm_stats(
    const float* __restrict__ sX, const float* __restrict__ tX,
    const float* __restrict__ sW, const float* __restrict__ tW,
    const int* __restrict__ tgt, float* __restrict__ partial)
{
    extern __shared__ float smem[];   // SMEM_FLOATS floats (dynamic LDS)

    const int tid  = threadIdx.x;
    const int lane = tid & 31;
    const int wid  = tid >> 5;
    const int wm   = wid >> 2, wn = wid & 3;
    const int half = lane >> 4, mr = lane & 15;
    const int rowBase = blockIdx.x * BM;
    const int colBase = blockIdx.y * BN;

    v8f sacc[2][2] = {};
    v8f tacc[2][2] = {};
    gemm_accum(sX, sW, HS, rowBase, colBase, smem, sacc);
    gemm_accum(tX, tW, HT, rowBase, colBase, smem, tacc);

    // Spill both 64x128 logit tiles to LDS (C/D layout: VGPR r -> row r + 8*half).
    float* S = smem;
    float* T = smem + BM * BN;
    #pragma unroll
    for (int mi = 0; mi < 2; ++mi)
        #pragma unroll
        for (int ni = 0; ni < 2; ++ni) {
            const int row0 = wm * 32 + mi * 16 + half * 8;
            const int col  = wn * 32 + ni * 16 + mr;
            #pragma unroll
            for (int r = 0; r < 8; ++r) {
                S[(row0 + r) * BN + col] = sacc[mi][ni][r];
                T[(row0 + r) * BN + col] = tacc[mi][ni][r];
            }
        }
    __syncthreads();

    // Row stats: 4 lanes per row, 32 cols each; combine via shuffles (xor 1,2).
    const int row = tid >> 2;            // 0..63
    const int q   = tid & 3;
    const float* srow = &S[row * BN + q * 32];
    const float* trow = &T[row * BN + q * 32];

    float m = -INFINITY;
    #pragma unroll 8
    for (int j = 0; j < 32; ++j) m = fmaxf(m, srow[j]);
    m = fmaxf(m, __shfl_xor(m, 1, 32));
    m = fmaxf(m, __shfl_xor(m, 2, 32));

    float e = 0.f, ss = 0.f, ttv = 0.f, st = 0.f;
    #pragma unroll 8
    for (int j = 0; j < 32; ++j) {
        const float sv = srow[j], tv = trow[j];
        e   += __expf(sv - m);
        ss  += sv * sv;
        ttv += tv * tv;
        st  += sv * tv;
    }
    const int tg = tgt[rowBase + row];
    float tl = 0.f;
    const int rel = tg - colBase - q * 32;
    if (rel >= 0 && rel < 32) tl = srow[rel];

    e   += __shfl_xor(e, 1, 32);   e   += __shfl_xor(e, 2, 32);
    ss  += __shfl_xor(ss, 1, 32);  ss  += __shfl_xor(ss, 2, 32);
    ttv += __shfl_xor(ttv, 1, 32); ttv += __shfl_xor(ttv, 2, 32);
    st  += __shfl_xor(st, 1, 32);  st  += __shfl_xor(st, 2, 32);
    tl  += __shfl_xor(tl, 1, 32);  tl  += __shfl_xor(tl, 2, 32);

    if (q == 0) {
        const size_t base = ((size_t)(rowBase + row) * NBLK + blockIdx.y) * 6;
        partial[base + 0] = m;
        partial[base + 1] = e;
        partial[base + 2] = ss;
        partial[base + 3] = ttv;
        partial[base + 4] = st;
        partial[base + 5] = tl;
    }
}

// ---------------------------------------------------------------------------
// Kernel 2: per-row merge of NBLK partials (online logsumexp) -> row loss.
// One wave per row; 8 rows per block.
// ---------------------------------------------------------------------------
__global__ __launch_bounds__(256) void k_row_reduce(
    const float* __restrict__ partial, const int* __restrict__ tgt,
    float* __restrict__ rowloss)
{
    const int lane = threadIdx.x & 31;
    const int wid  = threadIdx.x >> 5;
    const int row  = blockIdx.x * 8 + wid;
    const float* p = &partial[(size_t)row * NBLK * 6];

    float m = -INFINITY, l = 0.f, ss = 0.f, tt = 0.f, st = 0.f, tl = 0.f;
    for (int nb = lane; nb < NBLK; nb += 32) {
        const float* qq = &p[nb * 6];
        const float mi = qq[0], li = qq[1];
        const float nm = fmaxf(m, mi);
        l = l * __expf(m - nm) + li * __expf(mi - nm);
        m = nm;
        ss += qq[2]; tt += qq[3]; st += qq[4]; tl += qq[5];
    }
    #pragma unroll
    for (int off = 16; off > 0; off >>= 1) {
        const float m2 = __shfl_xor(m, off, 32);
        const float l2 = __shfl_xor(l, off, 32);
        const float nm = fmaxf(m, m2);
        l = l * __expf(m - nm) + l2 * __expf(m2 - nm);
        m = nm;
        ss += __shfl_xor(ss, off, 32);
        tt += __shfl_xor(tt, off, 32);
        st += __shfl_xor(st, off, 32);
        tl += __shfl_xor(tl, off, 32);
    }
    if (lane == 0) {
        const int tg = tgt[row];
        const float ce   = (tg != IGNORE_INDEX) ? (m + __logf(l) - tl) : 0.0f;
        const float sn   = fmaxf(sqrtf(ss), EPSN);
        const float tn   = fmaxf(sqrtf(tt), EPSN);
        const float csim = st / (sn * tn);
        // WEIGHT_HARD*ce + WEIGHT_SOFT*BETA*(1-cos), /BT applied at the end
        rowloss[row] = 0.5f * ce + 0.25f * (1.0f - csim);
    }
}

// ---------------------------------------------------------------------------
// Kernel 3: reduce 4096 row losses -> scalar.
// ---------------------------------------------------------------------------
__global__ __launch_bounds__(256) void k_final(
    const float* __restrict__ rowloss, float* __restrict__ out)
{
    __shared__ float red[256];
    float s = 0.f;
    for (int i = threadIdx.x; i < BT; i += 256) s += rowloss[i];
    red[threadIdx.x] = s;
    __syncthreads();
    #pragma unroll
    for (int stride = 128; stride > 0; stride >>= 1) {
        if (threadIdx.x < stride) red[threadIdx.x] += red[threadIdx.x + stride];
        __syncthreads();
    }
    if (threadIdx.x == 0) out[0] = red[0] / (float)BT;
}

// ---------------------------------------------------------------------------
extern "C" void kernel_launch(void* const* d_in, const int* in_sizes, int n_in,
                              void* d_out, int out_size, void* d_ws, size_t ws_size,
                              hipStream_t stream)
{
    const float* sX = (const float*)d_in[0];   // [BT, HS]
    const float* tX = (const float*)d_in[1];   // [BT, HT]
    const float* sW = (const float*)d_in[2];   // [V, HS]
    const float* tW = (const float*)d_in[3];   // [V, HT]
    const int*   tg = (const int*)d_in[4];     // [BT]
    float* out = (float*)d_out;

    float* partial = (float*)d_ws;                               // BT*NBLK*6 floats
    float* rowloss = partial + (size_t)BT * NBLK * 6;            // BT floats

    dim3 grid1(BT / BM, NBLK);   // M fastest -> M-tiles of one N-slab co-scheduled (L2 weight reuse)
    k_fused_gemm_stats<<<grid1, 256, SMEM_BYTES, stream>>>(sX, tX, sW, tW, tg, partial);
    k_row_reduce<<<BT / 8, 256, 0, stream>>>(partial, tg, rowloss);
    k_final<<<1, 256, 0, stream>>>(rowloss, out);
}